// GNN_90881507983447
// MI455X (gfx1250) — compile-verified
//
#include <hip/hip_runtime.h>
#include <hip/hip_bf16.h>

#define N_NODES 50000
#define N_EDGES 800000

typedef __attribute__((ext_vector_type(2))) float v2f;
typedef __attribute__((ext_vector_type(8))) float v8f;

// ---------------------------------------------------------------------------
// Edge scatter-add: one wave (32 lanes) per edge. Lane loads VEC contiguous
// floats from the source row (coalesced 512B/256B per wave; the 25.6MB feature
// table is L2-resident on the 192MB L2), then non-returning f32 atomics into
// the destination row. This phase dominates the roofline (~1.6GB total).
// ---------------------------------------------------------------------------
template <int VEC>
__global__ void gnn_scatter_add(const float* __restrict__ src,
                                float* __restrict__ dst,
                                const long long* __restrict__ row,
                                const long long* __restrict__ col) {
    const int d = VEC * 32;
    long long e = ((long long)blockIdx.x * blockDim.x + threadIdx.x) >> 5;
    int lane = threadIdx.x & 31;
    if (e >= N_EDGES) return;               // wave-uniform
    long long r = row[e];
    long long c = col[e];
    const float* s = src + c * d + lane * VEC;
    float* p = dst + r * d + lane * VEC;
    float v[VEC];
    if constexpr (VEC == 4) {
        float4 t = *(const float4*)s;
        v[0] = t.x; v[1] = t.y; v[2] = t.z; v[3] = t.w;
    } else {
        float2 t = *(const float2*)s;
        v[0] = t.x; v[1] = t.y;
    }
#pragma unroll
    for (int i = 0; i < VEC; ++i) atomicAdd(p + i, v[i]);
}

// ---------------------------------------------------------------------------
// Y[50000,64] = relu(X[50000,DIN] @ W[DIN,64] + b) with V_WMMA_F32_16X16X4_F32.
//
// W (<= 32KB) is identical for every wave, so it is staged ONCE per workgroup
// into LDS with the CDNA5 async global->LDS DMA path
// (global_load_async_to_lds_b128, tracked by ASYNCcnt), then all B fragments
// come from LDS -> no global W traffic in the K loop, and the only
// s_wait_loadcnt in steady state covers the streaming A loads.
//
// One wave owns a 16-row strip and all 64 output columns (4 C tiles of 16x16).
// A-matrix 16x4 f32 layout (ISA 7.12.2): lane m=lane&15 holds row m,
//   VGPR j holds K = (lane>=16 ? 2 : 0) + j.
// B-matrix 4x16: VGPR j, lane L -> K = (L>=16 ? 2 : 0)+j, N = L&15.
// C/D 16x16 f32: VGPR r -> M = r + 8*(lane>=16), N = lane&15.
// Wave-uniform exit (after the block-wide barrier) keeps EXEC all-ones.
// ---------------------------------------------------------------------------
template <int DIN>
__global__ __launch_bounds__(256)
void gnn_linear_relu_wmma(const float* __restrict__ X,
                          const float* __restrict__ W,
                          const float* __restrict__ bias,
                          float* __restrict__ Y) {
    __shared__ float lds_w[DIN * 64];       // 32KB (DIN=128) / 16KB (DIN=64)

    // ---- async DMA W into LDS: 16B per lane per issue ----
    constexpr int CHUNKS = DIN * 64 * 4 / 16;     // 16B chunks
    unsigned lds_base = (unsigned)(size_t)lds_w;  // low 32 bits = LDS offset
    int tid = threadIdx.x;
#pragma unroll
    for (int i = tid; i < CHUNKS; i += 256) {
        unsigned long long gaddr = (unsigned long long)(const void*)(W + i * 4);
        unsigned laddr = lds_base + (unsigned)i * 16u;
        asm volatile("global_load_async_to_lds_b128 %0, %1, off"
                     :: "v"(laddr), "v"(gaddr)
                     : "memory");
    }
    asm volatile("s_wait_asynccnt 0x0" ::: "memory");
    __syncthreads();

    int wave_id = (int)(((long long)blockIdx.x * blockDim.x + threadIdx.x) >> 5);
    int lane = threadIdx.x & 31;
    int tile = wave_id * 16;
    if (tile >= N_NODES) return;            // wave-uniform
    int m  = lane & 15;
    int kh = lane >> 4;                     // 0 or 1 (K half)

    v8f acc[4] = {v8f{0}, v8f{0}, v8f{0}, v8f{0}};
    const float* xrow = X + (long long)(tile + m) * DIN;

#pragma unroll 4
    for (int k0 = 0; k0 < DIN; k0 += 4) {
        int ka = k0 + kh * 2;
        v2f a;
        a.x = xrow[ka + 0];
        a.y = xrow[ka + 1];
        const float* wr0 = &lds_w[ka * 64];       // row ka   of W[DIN,64]
        const float* wr1 = wr0 + 64;              // row ka+1
#pragma unroll
        for (int t = 0; t < 4; ++t) {
            v2f b;
            b.x = wr0[t * 16 + m];
            b.y = wr1[t * 16 + m];
            acc[t] = __builtin_amdgcn_wmma_f32_16x16x4_f32(
                false, a, false, b, (short)0, acc[t], false, false);
        }
    }

#pragma unroll
    for (int t = 0; t < 4; ++t) {
        float bv = bias[t * 16 + m];
#pragma unroll
        for (int r = 0; r < 8; ++r) {
            int rowi = tile + r + 8 * kh;
            float v = acc[t][r] + bv;
            Y[(long long)rowi * 64 + t * 16 + m] = v > 0.f ? v : 0.f;
        }
    }
}

// ---------------------------------------------------------------------------
// Column sum over [N_NODES,64]: grid-stride with stride a multiple of 64, so
// each thread owns a fixed column; one atomic per thread at the end.
// ---------------------------------------------------------------------------
__global__ void gnn_colsum(const float* __restrict__ X, float* __restrict__ pooled) {
    long long tid = (long long)blockIdx.x * blockDim.x + threadIdx.x;
    long long stride = (long long)gridDim.x * blockDim.x;   // multiple of 64
    const long long total = (long long)N_NODES * 64;
    float s = 0.f;
    for (long long i = tid; i < total; i += stride) s += X[i];
    atomicAdd(&pooled[(int)(tid & 63)], s);
}

// ---------------------------------------------------------------------------
// Head: one wave32. Lane j computes h1[j] = relu(mean(X) @ Wf1 + bf1)[j],
// then shuffle-reduce h1 @ Wf2, sigmoid, lane 0 writes the scalar.
// ---------------------------------------------------------------------------
__global__ void gnn_head(const float* __restrict__ pooled,
                         const float* __restrict__ Wf1, const float* __restrict__ bf1,
                         const float* __restrict__ Wf2, const float* __restrict__ bf2,
                         float* __restrict__ out) {
    int j = threadIdx.x;                    // 0..31
    const float inv_n = 1.0f / (float)N_NODES;
    float h1 = bf1[j];
#pragma unroll 8
    for (int i = 0; i < 64; ++i) h1 += (pooled[i] * inv_n) * Wf1[i * 32 + j];
    h1 = fmaxf(h1, 0.f);
    float p = h1 * Wf2[j];
#pragma unroll
    for (int off = 16; off >= 1; off >>= 1) p += __shfl_down(p, off, 32);
    if (j == 0) out[0] = 1.0f / (1.0f + __expf(-(p + bf2[0])));
}

// ---------------------------------------------------------------------------
extern "C" void kernel_launch(void* const* d_in, const int* in_sizes, int n_in,
                              void* d_out, int out_size, void* d_ws, size_t ws_size,
                              hipStream_t stream) {
    const float*     x   = (const float*)d_in[0];
    const long long* ei  = (const long long*)d_in[1];   // int64 edge_index [2, E]
    const float* W1 = (const float*)d_in[2];  const float* b1 = (const float*)d_in[3];
    const float* W2 = (const float*)d_in[4];  const float* b2 = (const float*)d_in[5];
    const float* W3 = (const float*)d_in[6];  const float* b3 = (const float*)d_in[7];
    const float* Wf1 = (const float*)d_in[8]; const float* bf1 = (const float*)d_in[9];
    const float* Wf2 = (const float*)d_in[10];const float* bf2 = (const float*)d_in[11];

    const long long* row = ei;              // edge_index[0]
    const long long* col = ei + N_EDGES;    // edge_index[1]

    float* bufA   = (float*)d_ws;                            // [N, up to 128]
    float* bufB   = bufA + (size_t)N_NODES * 128;            // [N, up to 128]
    float* pooled = bufB + (size_t)N_NODES * 128;            // [64]

    const int edge_blocks = (N_EDGES + 7) / 8;               // 8 waves/block, 1 edge/wave
    const int lin_blocks  = (N_NODES / 16 + 7) / 8;          // 16 rows/wave, 8 waves/block

    // ---- Layer 1 (din = 128) ----
    hipMemsetAsync(bufB, 0, (size_t)N_NODES * 128 * sizeof(float), stream);
    gnn_scatter_add<4><<<edge_blocks, 256, 0, stream>>>(x, bufB, row, col);
    gnn_linear_relu_wmma<128><<<lin_blocks, 256, 0, stream>>>(bufB, W1, b1, bufA);

    // ---- Layer 2 (din = 64) ----
    hipMemsetAsync(bufB, 0, (size_t)N_NODES * 64 * sizeof(float), stream);
    gnn_scatter_add<2><<<edge_blocks, 256, 0, stream>>>(bufA, bufB, row, col);
    gnn_linear_relu_wmma<64><<<lin_blocks, 256, 0, stream>>>(bufB, W2, b2, bufA);

    // ---- Layer 3 (din = 64) ----
    hipMemsetAsync(bufB, 0, (size_t)N_NODES * 64 * sizeof(float), stream);
    gnn_scatter_add<2><<<edge_blocks, 256, 0, stream>>>(bufA, bufB, row, col);
    gnn_linear_relu_wmma<64><<<lin_blocks, 256, 0, stream>>>(bufB, W3, b3, bufA);

    // ---- Pool + MLP head ----
    hipMemsetAsync(pooled, 0, 64 * sizeof(float), stream);
    gnn_colsum<<<390, 256, 0, stream>>>(bufA, pooled);
    gnn_head<<<1, 32, 0, stream>>>(pooled, Wf1, bf1, Wf2, bf2, (float*)d_out);
}